// AdaptivePatchEncoder_82695300317515
// MI455X (gfx1250) — compile-verified
//
#include <hip/hip_runtime.h>
#include <cmath>

// ---------------------------------------------------------------------------
// AdaptivePatchEncoder for MI455X (gfx1250, wave32).
// - All dense GEMMs on the matrix pipe via V_WMMA_F32_16X16X4_F32 (exact f32;
//   workload is HBM-bound at ~48 GFLOP / ~140MB so f32 WMMA is both fast
//   enough and numerically faithful to the f32 reference).
// - GEMM A-blocks staged into LDS with GLOBAL_LOAD_ASYNC_TO_LDS_B128
//   (ASYNCcnt path), drained with s_wait_asynccnt before the barrier.
// ---------------------------------------------------------------------------

typedef float v2f __attribute__((ext_vector_type(2)));
typedef float v8f __attribute__((ext_vector_type(8)));

#define WMMA_F32(a, b, c) \
  __builtin_amdgcn_wmma_f32_16x16x4_f32(false, (a), false, (b), (short)0, (c), false, false)

// Problem constants
#define BB 32
#define PP 64
#define LL 2048      // points per trajectory
#define TT 32        // tokens per patch
#define DD 256
#define DFF 1024
#define NPATCH 2048  // B*P
#define NPOINT 65536 // B*L
#define LDA260 260   // 256 + 4 pad: 16B-aligned rows (async B128) + conflict-free frags

__device__ __forceinline__ float gelu_exact(float x) {
  return 0.5f * x * (1.0f + erff(x * 0.70710678118654752440f));
}

__device__ __forceinline__ float wave_sum32(float v) {
  for (int m = 16; m; m >>= 1) v += __shfl_xor(v, m, 32);
  return v;
}
__device__ __forceinline__ float wave_max32(float v) {
  for (int m = 16; m; m >>= 1) v = fmaxf(v, __shfl_xor(v, m, 32));
  return v;
}

// Async 16B copy global -> LDS (per-lane), tracked by ASYNCcnt.
__device__ __forceinline__ void async_copy_b128(unsigned lds_off, const float* gsrc) {
  asm volatile("global_load_async_to_lds_b128 %0, %1, off"
               :: "v"(lds_off), "v"((unsigned long long)(uintptr_t)gsrc)
               : "memory");
}
__device__ __forceinline__ void async_wait0() {
  asm volatile("s_wait_asynccnt 0" ::: "memory");
}

// ---------------------------------------------------------------------------
// Kernel 1: point MLP.  pt = gelu(feats @ W1 + b1) @ W2 + b2, * attn_mask.
// One WG = 16 point rows.  Hidden computed into LDS, second GEMM via WMMA.
// ---------------------------------------------------------------------------
__global__ __launch_bounds__(256) void point_mlp_kernel(
    const float* __restrict__ traj, const float* __restrict__ amask,
    const float* __restrict__ intervals, const float* __restrict__ omask,
    const float* __restrict__ W1, const float* __restrict__ b1,
    const float* __restrict__ W2, const float* __restrict__ b2,
    float* __restrict__ pt) {
  __shared__ __align__(16) float Hs[16 * LDA260];
  __shared__ float feats_s[16][4];
  __shared__ float mask_s[16];
  const int m0 = blockIdx.x * 16;
  if (threadIdx.x < 16) {
    const int row = m0 + threadIdx.x;
    feats_s[threadIdx.x][0] = traj[row * 2 + 0];
    feats_s[threadIdx.x][1] = traj[row * 2 + 1];
    feats_s[threadIdx.x][2] = intervals[row];
    feats_s[threadIdx.x][3] = omask[row];
    mask_s[threadIdx.x] = amask[row];
  }
  __syncthreads();
  // hidden = gelu(feats @ W1 + b1)   (K=4, trivial on VALU)
  for (int t = threadIdx.x; t < 16 * DD; t += 256) {
    const int r = t >> 8, j = t & 255;
    float acc = b1[j];
#pragma unroll
    for (int f = 0; f < 4; ++f) acc += feats_s[r][f] * W1[f * DD + j];
    Hs[r * LDA260 + j] = gelu_exact(acc);
  }
  __syncthreads();
  // pt = hidden @ W2 + b2   (WMMA f32 16x16x4, W2 is [K,N] row-major)
  const int wave = threadIdx.x >> 5, lane = threadIdx.x & 31;
  const int half = lane >> 4, sl = lane & 15, kb = half * 2;
  for (int nt = wave; nt < DD / 16; nt += 8) {
    const int n0 = nt * 16;
    v8f acc = {};
    for (int k = 0; k < DD; k += 4) {
      v2f a = *(const v2f*)&Hs[sl * LDA260 + k + kb];
      v2f b;
      b.x = W2[(size_t)(k + kb) * DD + n0 + sl];
      b.y = W2[(size_t)(k + kb + 1) * DD + n0 + sl];
      acc = WMMA_F32(a, b, acc);
    }
    const int n = n0 + sl;
    const float bv = b2[n];
#pragma unroll
    for (int r = 0; r < 8; ++r) {
      const int mr = r + half * 8;
      pt[(size_t)(m0 + mr) * DD + n] = (acc[r] + bv) * mask_s[mr];
    }
  }
}

// ---------------------------------------------------------------------------
// Kernel 2: masked pooling + gather.  One WG per patch.
// ---------------------------------------------------------------------------
__global__ __launch_bounds__(256) void pool_gather_kernel(
    const float* __restrict__ p2p, const float* __restrict__ amask,
    const float* __restrict__ pt, const float* __restrict__ len_emb,
    float* __restrict__ q0, int* __restrict__ idx_out, int* __restrict__ len_out,
    float* __restrict__ patch_pad, float* __restrict__ lengths_f) {
  __shared__ unsigned char flags[LL];
  __shared__ int idx_s[TT];
  __shared__ int count_s;
  const int n = blockIdx.x;        // patch id b*P + p
  const int b = n >> 6;            // P = 64
  const float* vrow = p2p + (size_t)n * LL;
  const float* arow = amask + (size_t)b * LL;
  for (int l = threadIdx.x; l < LL; l += 256)
    flags[l] = (vrow[l] * arow[l]) > 0.5f ? 1 : 0;
  __syncthreads();
  if (threadIdx.x == 0) {
    int c = 0;
    for (int l = 0; l < LL; ++l)
      if (flags[l]) { if (c < TT) idx_s[c] = l; ++c; }
    count_s = c;
  }
  __syncthreads();
  const int cnt = count_s;
  const int g = cnt < TT ? cnt : TT;
  const int d = threadIdx.x;
  float s = 0.f;
  for (int j = 0; j < g; ++j)
    s += pt[((size_t)b * LL + idx_s[j]) * DD + d];
  const float pooled = s / fmaxf((float)cnt, 1.f);
  const int ce = cnt < 64 ? cnt : 64;
  q0[(size_t)n * DD + d] = pooled + len_emb[(size_t)ce * DD + d];
  if (d < TT) idx_out[n * TT + d] = (d < g) ? idx_s[d] : 0;
  if (d == 0) {
    len_out[n] = cnt;
    patch_pad[n] = (cnt <= 0) ? 1.f : 0.f;
    lengths_f[n] = (float)cnt;
  }
}

// ---------------------------------------------------------------------------
// Row LayerNorm over D=256 (one row per wave32, 8 rows per WG).
// ---------------------------------------------------------------------------
__global__ __launch_bounds__(256) void ln_rows_kernel(
    const float* __restrict__ X, float* __restrict__ Y,
    const float* __restrict__ g, const float* __restrict__ b) {
  const int wave = threadIdx.x >> 5, lane = threadIdx.x & 31;
  const int row = blockIdx.x * 8 + wave;
  const float* x = X + (size_t)row * DD;
  float v[8];
  float s = 0.f;
#pragma unroll
  for (int i = 0; i < 8; ++i) { v[i] = x[lane + 32 * i]; s += v[i]; }
  s = wave_sum32(s);
  const float mean = s * (1.f / 256.f);
  float vs = 0.f;
#pragma unroll
  for (int i = 0; i < 8; ++i) { const float dd = v[i] - mean; vs += dd * dd; }
  vs = wave_sum32(vs);
  const float rinv = rsqrtf(vs * (1.f / 256.f) + 1e-5f);
#pragma unroll
  for (int i = 0; i < 8; ++i) {
    const int c = lane + 32 * i;
    Y[(size_t)row * DD + c] = (v[i] - mean) * rinv * g[c] + b[c];
  }
}

// ---------------------------------------------------------------------------
// Batched GEMM:  C[M,N] = epi(A[M,K] @ W (+bias) [, residual R]).
// BT=true : W is [N,K] row-major, computes A @ W^T   (q/o projections)
// BT=false: W is [K,N] row-major, computes A @ W     (FFN)
// One WG = 16 rows of A, staged memory->LDS with async B128 copies
// (ASYNCcnt); 8 waves sweep N in 16-wide WMMA tiles.
// ---------------------------------------------------------------------------
enum { EPI_BIAS = 0, EPI_GELU = 1, EPI_RES = 2 };

template <int K, int N, bool BT, int EPI>
__global__ __launch_bounds__(256) void gemm_kernel(
    const float* __restrict__ A, const float* __restrict__ W,
    const float* __restrict__ bias, const float* __restrict__ R,
    float* __restrict__ C) {
  constexpr int LDA = K + 4;  // rows stay 16B aligned for async B128
  __shared__ __align__(16) float As[16 * LDA];
  const int m0 = blockIdx.x * 16;
  // ---- async stage A block (16 x K) into LDS: 16B chunks, no VGPR bounce ----
  {
    const unsigned lds_base = (unsigned)(uintptr_t)&As[0];
    constexpr int CHUNKS = 16 * (K / 4);  // multiples of 256 -> EXEC-uniform
    for (int t = threadIdx.x; t < CHUNKS; t += 256) {
      const int r = t / (K / 4), c = t - r * (K / 4);
      async_copy_b128(lds_base + (unsigned)((r * LDA + c * 4) * sizeof(float)),
                      A + (size_t)(m0 + r) * K + c * 4);
    }
    async_wait0();
  }
  __syncthreads();
  const int wave = threadIdx.x >> 5, lane = threadIdx.x & 31;
  const int half = lane >> 4, sl = lane & 15, kb = half * 2;
  for (int nt = wave; nt < N / 16; nt += 8) {
    const int n0 = nt * 16;
    if (BT) __builtin_prefetch(&W[(size_t)(n0 + sl) * K], 0, 1);
    v8f acc = {};
#pragma unroll 4
    for (int k = 0; k < K; k += 4) {
      v2f a = *(const v2f*)&As[sl * LDA + k + kb];
      v2f b;
      if (BT) {
        b = *(const v2f*)&W[(size_t)(n0 + sl) * K + k + kb];
      } else {
        b.x = W[(size_t)(k + kb) * N + n0 + sl];
        b.y = W[(size_t)(k + kb + 1) * N + n0 + sl];
      }
      acc = WMMA_F32(a, b, acc);
    }
    const int n = n0 + sl;
    const float bv = bias[n];
#pragma unroll
    for (int r = 0; r < 8; ++r) {
      const int m = m0 + r + half * 8;
      float v = acc[r] + bv;
      if (EPI == EPI_GELU) v = gelu_exact(v);
      if (EPI == EPI_RES) v += R[(size_t)m * N + n];
      C[(size_t)m * N + n] = v;
    }
  }
}

// ---------------------------------------------------------------------------
// Per-patch cross attention core.  One WG per patch (8 waves).
// gather kv -> LN -> kh=kvn@Wk^T (WMMA) -> masked softmax (one head per
// wave32, one key per lane) -> vh=kvn@Wv^T (WMMA, reusing kh buffer) -> o.
// ---------------------------------------------------------------------------
__device__ __forceinline__ void tile_proj_256(
    const float* __restrict__ Asrc, const float* __restrict__ W, int mt, int nt,
    const float* __restrict__ bias, float* __restrict__ dst, int lane) {
  const int half = lane >> 4, sl = lane & 15, kb = half * 2;
  v8f acc = {};
  for (int k = 0; k < DD; k += 4) {
    v2f a = *(const v2f*)&Asrc[(mt * 16 + sl) * LDA260 + k + kb];
    v2f b = *(const v2f*)&W[(size_t)(nt * 16 + sl) * DD + k + kb];
    acc = WMMA_F32(a, b, acc);
  }
  const int n = nt * 16 + sl;
  const float bv = bias[n];
#pragma unroll
  for (int r = 0; r < 8; ++r)
    dst[(mt * 16 + r + half * 8) * LDA260 + n] = acc[r] + bv;
}

__global__ __launch_bounds__(256) void patch_attn_kernel(
    const float* __restrict__ qh, const float* __restrict__ pt,
    const int* __restrict__ idx, const int* __restrict__ lens,
    const float* __restrict__ lnkv_g, const float* __restrict__ lnkv_b,
    const float* __restrict__ Wk, const float* __restrict__ bk,
    const float* __restrict__ Wv, const float* __restrict__ bv,
    float* __restrict__ o_out) {
  __shared__ __align__(16) float kv[TT * LDA260];   // kvn (after LN)
  __shared__ __align__(16) float khv[TT * LDA260];  // kh, then vh
  __shared__ float prob[8][TT];
  __shared__ float qh_s[DD];
  const int n = blockIdx.x, b = n >> 6;
  const int len = lens[n];
  const int wave = threadIdx.x >> 5, lane = threadIdx.x & 31;
  qh_s[threadIdx.x] = qh[(size_t)n * DD + threadIdx.x];
  // gather (padded rows zeroed, as in the reference seqs buffer)
  for (int r = 0; r < TT; ++r) {
    float v = 0.f;
    if (r < len) v = pt[((size_t)b * LL + idx[n * TT + r]) * DD + threadIdx.x];
    kv[r * LDA260 + threadIdx.x] = v;
  }
  __syncthreads();
  // LayerNorm each kv row (wave w owns rows w, w+8, w+16, w+24)
  for (int r = wave; r < TT; r += 8) {
    float x[8];
    float s = 0.f;
#pragma unroll
    for (int i = 0; i < 8; ++i) { x[i] = kv[r * LDA260 + lane + 32 * i]; s += x[i]; }
    s = wave_sum32(s);
    const float mean = s * (1.f / 256.f);
    float vs = 0.f;
#pragma unroll
    for (int i = 0; i < 8; ++i) { const float dd = x[i] - mean; vs += dd * dd; }
    vs = wave_sum32(vs);
    const float rinv = rsqrtf(vs * (1.f / 256.f) + 1e-5f);
#pragma unroll
    for (int i = 0; i < 8; ++i) {
      const int c = lane + 32 * i;
      kv[r * LDA260 + c] = (x[i] - mean) * rinv * lnkv_g[c] + lnkv_b[c];
    }
  }
  __syncthreads();
  // kh = kvn @ Wk^T + bk   (2 m-tiles x 16 n-tiles = 32 tiles / 8 waves)
  for (int t = wave; t < 32; t += 8)
    tile_proj_256(kv, Wk, t >> 4, t & 15, bk, khv, lane);
  __syncthreads();
  // scores + softmax: wave = head, lane = key
  {
    const int h = wave, k = lane;
    float sc = 0.f;
#pragma unroll 8
    for (int i = 0; i < 32; ++i) sc += qh_s[h * 32 + i] * khv[k * LDA260 + h * 32 + i];
    sc *= 0.17677669529663687f;  // 1/sqrt(32)
    const bool pad = (k >= len) && !(len == 0 && k == 0);  // kpm_safe
    if (pad) sc = -1e30f;
    const float mx = wave_max32(sc);
    const float e = expf(sc - mx);
    const float se = wave_sum32(e);
    prob[h][k] = e / se;
  }
  __syncthreads();
  // vh = kvn @ Wv^T + bv  (overwrites khv)
  for (int t = wave; t < 32; t += 8)
    tile_proj_256(kv, Wv, t >> 4, t & 15, bv, khv, lane);
  __syncthreads();
  // o[d] = sum_k prob[d/32][k] * vh[k][d]
  {
    const int d = threadIdx.x, h = d >> 5;
    float s = 0.f;
#pragma unroll 8
    for (int k = 0; k < TT; ++k) s += prob[h][k] * khv[k * LDA260 + d];
    o_out[(size_t)n * DD + d] = s;
  }
}

// ---------------------------------------------------------------------------
// Finalize: out_tokens = q * ~patch_pad
// ---------------------------------------------------------------------------
__global__ __launch_bounds__(256) void finalize_kernel(
    const float* __restrict__ q, const int* __restrict__ lens,
    float* __restrict__ out_tokens) {
  const int i = blockIdx.x * 256 + threadIdx.x;
  const int n = i >> 8;
  out_tokens[i] = (lens[n] > 0) ? q[i] : 0.f;
}

// ---------------------------------------------------------------------------
// Host orchestration.
// Input order (setup_inputs dict order; nested `layers` dict expanded in its
// own insertion order): 0 trajectory, 1 attention_mask, 2 patch2point_mask,
// 3 intervals, 4 observed_mask, 5 mlp_W1, 6 mlp_b1, 7 mlp_W2, 8 mlp_b2,
// 9 len_emb, then layers: 10 lnq_g, 11 lnq_b, 12 lnkv_g, 13 lnkv_b, 14 Wq,
// 15 bq, 16 Wk, 17 bk, 18 Wv, 19 bv, 20 Wo, 21 bo, 22 lno_g, 23 lno_b,
// 24 W1, 25 b1, 26 W2, 27 b2.
// Output layout: out_tokens[524288] | patch_pad[2048] | pt[16777216] | lengths[2048]
// ---------------------------------------------------------------------------
extern "C" void kernel_launch(void* const* d_in, const int* in_sizes, int n_in,
                              void* d_out, int out_size, void* d_ws, size_t ws_size,
                              hipStream_t stream) {
  (void)in_sizes; (void)n_in; (void)out_size; (void)ws_size;
  const float* traj      = (const float*)d_in[0];
  const float* amask     = (const float*)d_in[1];
  const float* p2p       = (const float*)d_in[2];
  const float* intervals = (const float*)d_in[3];
  const float* omask     = (const float*)d_in[4];
  const float* mlpW1     = (const float*)d_in[5];
  const float* mlpb1     = (const float*)d_in[6];
  const float* mlpW2     = (const float*)d_in[7];
  const float* mlpb2     = (const float*)d_in[8];
  const float* len_emb   = (const float*)d_in[9];
  const float* lnq_g  = (const float*)d_in[10];
  const float* lnq_b  = (const float*)d_in[11];
  const float* lnkv_g = (const float*)d_in[12];
  const float* lnkv_b = (const float*)d_in[13];
  const float* Wq     = (const float*)d_in[14];
  const float* bq     = (const float*)d_in[15];
  const float* Wk     = (const float*)d_in[16];
  const float* bk     = (const float*)d_in[17];
  const float* Wv     = (const float*)d_in[18];
  const float* bvp    = (const float*)d_in[19];
  const float* Wo     = (const float*)d_in[20];
  const float* bo     = (const float*)d_in[21];
  const float* lno_g  = (const float*)d_in[22];
  const float* lno_b  = (const float*)d_in[23];
  const float* W1     = (const float*)d_in[24];
  const float* b1     = (const float*)d_in[25];
  const float* W2     = (const float*)d_in[26];
  const float* b2     = (const float*)d_in[27];

  float* out        = (float*)d_out;
  float* out_tokens = out;
  float* patch_pad  = out + 524288;
  float* pt         = out + 526336;
  float* lengths_f  = out + 17303552;

  float* ws = (float*)d_ws;
  float* q  = ws;
  float* qn = q  + (size_t)NPATCH * DD;
  float* qh = qn + (size_t)NPATCH * DD;
  float* o  = qh + (size_t)NPATCH * DD;
  float* h  = o  + (size_t)NPATCH * DD;
  float* hn = h  + (size_t)NPATCH * DD;
  float* f1 = hn + (size_t)NPATCH * DD;
  int* idx  = (int*)(f1 + (size_t)NPATCH * DFF);
  int* lens = idx + (size_t)NPATCH * TT;

  point_mlp_kernel<<<NPOINT / 16, 256, 0, stream>>>(
      traj, amask, intervals, omask, mlpW1, mlpb1, mlpW2, mlpb2, pt);
  pool_gather_kernel<<<NPATCH, 256, 0, stream>>>(
      p2p, amask, pt, len_emb, q, idx, lens, patch_pad, lengths_f);

  for (int i = 0; i < 2; ++i) {
    const size_t wOff  = (size_t)i * DD * DD;
    const size_t vOff  = (size_t)i * DD;
    const size_t w1Off = (size_t)i * DD * DFF;
    const size_t b1Off = (size_t)i * DFF;
    // qn = LN(q); qh = qn @ Wq^T + bq
    ln_rows_kernel<<<NPATCH / 8, 256, 0, stream>>>(q, qn, lnq_g + vOff, lnq_b + vOff);
    gemm_kernel<DD, DD, true, EPI_BIAS><<<NPATCH / 16, 256, 0, stream>>>(
        qn, Wq + wOff, bq + vOff, nullptr, qh);
    // per-patch attention core -> o
    patch_attn_kernel<<<NPATCH, 256, 0, stream>>>(
        qh, pt, idx, lens, lnkv_g + vOff, lnkv_b + vOff,
        Wk + wOff, bk + vOff, Wv + wOff, bvp + vOff, o);
    // h = q + o @ Wo^T + bo
    gemm_kernel<DD, DD, true, EPI_RES><<<NPATCH / 16, 256, 0, stream>>>(
        o, Wo + wOff, bo + vOff, q, h);
    // hn = LN(h); f1 = gelu(hn @ W1 + b1); q = h + f1 @ W2 + b2
    ln_rows_kernel<<<NPATCH / 8, 256, 0, stream>>>(h, hn, lno_g + vOff, lno_b + vOff);
    gemm_kernel<DD, DFF, false, EPI_GELU><<<NPATCH / 16, 256, 0, stream>>>(
        hn, W1 + w1Off, b1 + b1Off, nullptr, f1);
    gemm_kernel<DFF, DD, false, EPI_RES><<<NPATCH / 16, 256, 0, stream>>>(
        f1, W2 + w1Off, b2 + vOff, h, q);
  }

  finalize_kernel<<<NPATCH, 256, 0, stream>>>(q, lens, out_tokens);
}